// AGNN_EFG_10557029614293
// MI455X (gfx1250) — compile-verified
//
#include <hip/hip_runtime.h>
#include <hip/hip_bf16.h>

#define N_TOT 131072
#define EDGES 2097152
#define HID   64
#define NB    64
#define N_PER 2048
#define KEEP1 1024
#define NODES2 (NB * KEEP1)   // 65536
#define KEEP2 512
#define NODES3 (NB * KEEP2)   // 32768

typedef __attribute__((ext_vector_type(2))) float v2f;
typedef __attribute__((ext_vector_type(8))) float v8f;

__device__ __forceinline__ float leakyf(float v, float a) { return v > 0.f ? v : a * v; }

__device__ __forceinline__ void atomicMaxF(float* addr, float val) {
    if (val >= 0.f) atomicMax((int*)addr, __float_as_int(val));
    else            atomicMin((unsigned int*)addr, __float_as_uint(val));
}

// ---------------------------------------------------------------------------
// WMMA fp32 GEMM: Y = pre(X) @ W (+bias), optionally Y2 = pre(X) @ W2.
// X is [M x 64] row-major, W [64 x 64] row-major. M multiple of 64.
// PRE_BN applies t = leaky((x + b1 - mean)*rsqrt(var+eps)*gamma + beta, .01)
// per column while staging X into LDS.
// One 128-thread block = 4 waves, each wave computes a 16x64 tile via
// V_WMMA_F32_16X16X4_F32 (16 K-steps of 4). DUAL/PRE_BN are compile-time so
// the inner WMMA loop is branch-free and non-dual variants drop 32 VGPRs +
// 17KB LDS.
// ---------------------------------------------------------------------------
#define LDST 68   // padded LDS row stride (floats): bank-conflict-free fragments

template <bool DUAL, bool PRE_BN>
__global__ __launch_bounds__(128) void k_gemm64(
    const float* __restrict__ X, const float* __restrict__ W,
    const float* __restrict__ bias, float* __restrict__ Y,
    const float* __restrict__ W2, float* __restrict__ Y2,
    int M,
    const float* __restrict__ pb, const float* __restrict__ pg,
    const float* __restrict__ pbeta, const float* __restrict__ pm,
    const float* __restrict__ pv)
{
    __shared__ __align__(16) float sX [64 * LDST];
    __shared__ __align__(16) float sW [64 * LDST];                 // sW[n*LDST+k] = W[k][n]
    __shared__ __align__(16) float sW2[DUAL ? 64 * LDST : 4];

    const int tid  = threadIdx.x;
    const int wave = tid >> 5, lane = tid & 31;
    const int l16  = lane & 15, hb = lane >> 4;   // hb: half-wave select
    const int blockRow = blockIdx.x * 64;

    for (int i = tid; i < 4096; i += 128) {
        int k = i >> 6, n = i & 63;
        sW[n * LDST + k] = W[i];
        if (DUAL) sW2[n * LDST + k] = W2[i];
    }
    for (int i = tid; i < 64 * 16; i += 128) {
        int r  = i >> 4;
        int c4 = (i & 15) << 2;
        float4 v = *(const float4*)&X[(size_t)(blockRow + r) * HID + c4];
        if (PRE_BN) {
            float4 bb = *(const float4*)&pb[c4];
            float4 gg = *(const float4*)&pg[c4];
            float4 be = *(const float4*)&pbeta[c4];
            float4 mm = *(const float4*)&pm[c4];
            float4 vv = *(const float4*)&pv[c4];
#define BNC(c) { float t = (v.c + bb.c - mm.c) * rsqrtf(vv.c + 1e-5f) * gg.c + be.c; v.c = t > 0.f ? t : 0.01f * t; }
            BNC(x) BNC(y) BNC(z) BNC(w)
#undef BNC
        }
        *(float4*)&sX[r * LDST + c4] = v;
    }
    __syncthreads();

    v8f acc0 = {}, acc1 = {}, acc2 = {}, acc3 = {};
    v8f acd0 = {}, acd1 = {}, acd2 = {}, acd3 = {};

    // A frag lane L: M = L%16, a.x = A[M][k0 + 2*(L/16)], a.y = +1
    // B frag lane L: N = L%16, b.x = B[k0 + 2*(L/16)][N]  (read from transposed LDS)
    const float* aB  = &sX [(wave * 16 + l16) * LDST + 2 * hb];
    const float* bB  = &sW [l16 * LDST + 2 * hb];
    const float* bB2 = &sW2[l16 * LDST + 2 * hb];

#pragma unroll
    for (int k0 = 0; k0 < 64; k0 += 4) {
        v2f a  = *(const v2f*)(aB + k0);
        v2f b0 = *(const v2f*)(bB + k0);
        v2f b1 = *(const v2f*)(bB + 16 * LDST + k0);
        v2f b2 = *(const v2f*)(bB + 32 * LDST + k0);
        v2f b3 = *(const v2f*)(bB + 48 * LDST + k0);
        acc0 = __builtin_amdgcn_wmma_f32_16x16x4_f32(false, a, false, b0, (short)0, acc0, false, false);
        acc1 = __builtin_amdgcn_wmma_f32_16x16x4_f32(false, a, false, b1, (short)0, acc1, false, false);
        acc2 = __builtin_amdgcn_wmma_f32_16x16x4_f32(false, a, false, b2, (short)0, acc2, false, false);
        acc3 = __builtin_amdgcn_wmma_f32_16x16x4_f32(false, a, false, b3, (short)0, acc3, false, false);
        if (DUAL) {
            v2f c0 = *(const v2f*)(bB2 + k0);
            v2f c1 = *(const v2f*)(bB2 + 16 * LDST + k0);
            v2f c2 = *(const v2f*)(bB2 + 32 * LDST + k0);
            v2f c3 = *(const v2f*)(bB2 + 48 * LDST + k0);
            acd0 = __builtin_amdgcn_wmma_f32_16x16x4_f32(false, a, false, c0, (short)0, acd0, false, false);
            acd1 = __builtin_amdgcn_wmma_f32_16x16x4_f32(false, a, false, c1, (short)0, acd1, false, false);
            acd2 = __builtin_amdgcn_wmma_f32_16x16x4_f32(false, a, false, c2, (short)0, acd2, false, false);
            acd3 = __builtin_amdgcn_wmma_f32_16x16x4_f32(false, a, false, c3, (short)0, acd3, false, false);
        }
    }

    float bi0 = 0.f, bi1 = 0.f, bi2 = 0.f, bi3 = 0.f;
    if (bias) { bi0 = bias[l16]; bi1 = bias[l16 + 16]; bi2 = bias[l16 + 32]; bi3 = bias[l16 + 48]; }
    // D layout: VGPR v -> row = v + 8*hb, col = 16*n + l16
    const int rb = blockRow + wave * 16 + 8 * hb;
#pragma unroll
    for (int v = 0; v < 8; ++v) {
        float* yr = &Y[(size_t)(rb + v) * HID];
        yr[l16]      = acc0[v] + bi0;
        yr[l16 + 16] = acc1[v] + bi1;
        yr[l16 + 32] = acc2[v] + bi2;
        yr[l16 + 48] = acc3[v] + bi3;
        if (DUAL) {
            float* y2r = &Y2[(size_t)(rb + v) * HID];
            y2r[l16]      = acd0[v];
            y2r[l16 + 16] = acd1[v];
            y2r[l16 + 32] = acd2[v];
            y2r[l16 + 48] = acd3[v];
        }
    }
}

// --------------------------- GCN helpers ----------------------------------
__global__ void k_fill1(float* p, int n) {
    int i = blockIdx.x * blockDim.x + threadIdx.x;
    if (i < n) p[i] = 1.0f;   // self-loop contribution to degree
}
__global__ void k_deg_scatter(const int* __restrict__ dst, float* __restrict__ deg) {
    int e = blockIdx.x * blockDim.x + threadIdx.x;
    if (e < EDGES) atomicAdd(&deg[dst[e]], 1.0f);
}
__global__ void k_rsqrt_inplace(float* p, int n) {
    int i = blockIdx.x * blockDim.x + threadIdx.x;
    if (i < n) p[i] = rsqrtf(fmaxf(p[i], 1.0f));
}
__global__ void k_edge_norm(const int* __restrict__ src, const int* __restrict__ dst,
                            const float* __restrict__ dinv, float* __restrict__ enorm) {
    int e = blockIdx.x * blockDim.x + threadIdx.x;
    if (e < EDGES) enorm[e] = dinv[src[e]] * dinv[dst[e]];
}
__global__ void k_self_init(const float* __restrict__ xw, const float* __restrict__ dinv,
                            float* __restrict__ h) {
    int t = blockIdx.x * blockDim.x + threadIdx.x;  // N_TOT*HID threads
    int i = t >> 6;
    float di = dinv[i];
    h[t] = xw[t] * di * di;
}
__global__ void k_gcn_scatter(const int* __restrict__ src, const int* __restrict__ dst,
                              const float* __restrict__ enorm, const float* __restrict__ xw,
                              float* __restrict__ h) {
    int t = blockIdx.x * blockDim.x + threadIdx.x;  // EDGES*HID threads
    int e = t >> 6, c = t & 63;
    atomicAdd(&h[(size_t)dst[e] * HID + c], xw[(size_t)src[e] * HID + c] * enorm[e]);
}

// --------------------------- TopK pooling ---------------------------------
// One block per graph; exact rank selection (rank r < k -> output slot r),
// matching lax.top_k descending order with lowest-index tie break.
__global__ __launch_bounds__(256) void k_topk(
    const float* __restrict__ Xn, const float* __restrict__ p,
    float* __restrict__ Yp, int* __restrict__ newId, int nPer, int kKeep)
{
    __shared__ float sS[2048];
    __shared__ int   sSlot[1024];
    __shared__ float sPn[64];
    __shared__ float sRed[64];
    const int g = blockIdx.x, tid = threadIdx.x;

    if (tid < 64) sRed[tid] = p[tid] * p[tid];
    __syncthreads();
    if (tid == 0) { float s = 0; for (int c = 0; c < 64; ++c) s += sRed[c]; sRed[0] = rsqrtf(s); }
    __syncthreads();
    if (tid < 64) sPn[tid] = p[tid] * sRed[0];
    __syncthreads();

    for (int i = tid; i < nPer; i += 256) {
        const float* row = &Xn[(size_t)(g * nPer + i) * HID];
        float s = 0;
        for (int c = 0; c < HID; ++c) s += row[c] * sPn[c];
        sS[i] = s;
    }
    __syncthreads();

    for (int i = tid; i < nPer; i += 256) {
        float si = sS[i];
        int r = 0;
        for (int j = 0; j < nPer; ++j) {
            float sj = sS[j];
            r += (sj > si) || (sj == si && j < i);
        }
        if (r < kKeep) sSlot[r] = i;
        if (newId) newId[g * nPer + i] = (r < kKeep) ? (g * kKeep + r) : -1;
    }
    __syncthreads();

    for (int idx = tid; idx < kKeep * HID; idx += 256) {
        int r = idx >> 6, c = idx & 63;
        int i = sSlot[r];
        Yp[(size_t)(g * kKeep + r) * HID + c] =
            Xn[(size_t)(g * nPer + i) * HID + c] * tanhf(sS[i]);
    }
}

// --------------------------- GATv2 ----------------------------------------
__device__ __forceinline__ float attdot(const float* a, const float* b,
                                        const float* __restrict__ att) {
    float s = 0;
    for (int c = 0; c < HID; ++c) { float v = a[c] + b[c]; s += leakyf(v, 0.2f) * att[c]; }
    return s;
}
__global__ void k_gat_self(const float* __restrict__ xl, const float* __restrict__ xr,
                           const float* __restrict__ att, float* __restrict__ es,
                           float* __restrict__ m) {
    int i = blockIdx.x * blockDim.x + threadIdx.x;
    if (i >= NODES2) return;
    float s = attdot(&xl[(size_t)i * HID], &xr[(size_t)i * HID], att);
    es[i] = s; m[i] = s;   // self-loop always present -> valid init for segment max
}
__global__ void k_gat_edge_max(const int* __restrict__ src, const int* __restrict__ dst,
                               const int* __restrict__ newId,
                               const float* __restrict__ xl, const float* __restrict__ xr,
                               const float* __restrict__ att,
                               int* __restrict__ s2, int* __restrict__ d2,
                               float* __restrict__ ee, float* __restrict__ m) {
    int e = blockIdx.x * blockDim.x + threadIdx.x;
    if (e >= EDGES) return;
    int ns = newId[src[e]], nd = newId[dst[e]];
    if (ns >= 0 && nd >= 0) {
        float s = attdot(&xl[(size_t)ns * HID], &xr[(size_t)nd * HID], att);
        s2[e] = ns; d2[e] = nd; ee[e] = s;
        atomicMaxF(&m[nd], s);
    } else {
        s2[e] = -1; d2[e] = -1; ee[e] = 0.f;   // pruned edge: alpha == 0 exactly
    }
}
__global__ void k_gat_selfsum(const float* __restrict__ es, const float* __restrict__ m,
                              float* __restrict__ ssum) {
    int i = blockIdx.x * blockDim.x + threadIdx.x;
    if (i < NODES2) ssum[i] = expf(es[i] - m[i]);
}
__global__ void k_gat_edge_exp(const int* __restrict__ d2, float* __restrict__ ee,
                               const float* __restrict__ m, float* __restrict__ ssum) {
    int e = blockIdx.x * blockDim.x + threadIdx.x;
    if (e >= EDGES) return;
    int d = d2[e];
    if (d >= 0) {
        float x = expf(ee[e] - m[d]);
        ee[e] = x;
        atomicAdd(&ssum[d], x);
    }
}
__global__ void k_gat_selfapply(const float* __restrict__ es, const float* __restrict__ m,
                                const float* __restrict__ ssum, const float* __restrict__ xl,
                                float* __restrict__ h2) {
    int t = blockIdx.x * blockDim.x + threadIdx.x;  // NODES2*HID
    int i = t >> 6;
    float alpha = expf(es[i] - m[i]) / (ssum[i] + 1e-16f);
    h2[t] = alpha * xl[t];
}
__global__ void k_gat_edge_apply(const int* __restrict__ s2, const int* __restrict__ d2,
                                 const float* __restrict__ ex, const float* __restrict__ ssum,
                                 const float* __restrict__ xl, float* __restrict__ h2) {
    int t = blockIdx.x * blockDim.x + threadIdx.x;  // EDGES*HID
    int e = t >> 6, c = t & 63;
    int d = d2[e];
    if (d >= 0) {
        float alpha = ex[e] / (ssum[d] + 1e-16f);
        atomicAdd(&h2[(size_t)d * HID + c], alpha * xl[(size_t)s2[e] * HID + c]);
    }
}
__global__ void k_gat_final(float* __restrict__ h2, const float* __restrict__ bgat) {
    int t = blockIdx.x * blockDim.x + threadIdx.x;  // NODES2*HID
    int c = t & 63;
    h2[t] = leakyf(h2[t] + bgat[c], 0.01f);
}

// --------------------------- pooled head ----------------------------------
__global__ __launch_bounds__(64) void k_head(const float* __restrict__ h4,
                                             const float* __restrict__ W3,
                                             const float* __restrict__ b3,
                                             float* __restrict__ out, int k2) {
    __shared__ float red[64];
    int g = blockIdx.x, c = threadIdx.x;
    const float* base = &h4[(size_t)(g * k2) * HID + c];
    float acc = 0;
    for (int r = 0; r < k2; ++r) acc += base[(size_t)r * HID];
    red[c] = acc * W3[c];
    __syncthreads();
    if (c == 0) {
        float s = 0;
        for (int i = 0; i < 64; ++i) s += red[i];
        out[g] = s + b3[0];
    }
}

// ---------------------------------------------------------------------------
extern "C" void kernel_launch(void* const* d_in, const int* in_sizes, int n_in,
                              void* d_out, int out_size, void* d_ws, size_t ws_size,
                              hipStream_t stream) {
    const float* x    = (const float*)d_in[0];
    const int*   ei   = (const int*)d_in[1];
    const int*   src  = ei;
    const int*   dst  = ei + EDGES;
    const float* W1   = (const float*)d_in[3];
    const float* b1   = (const float*)d_in[4];
    const float* bng  = (const float*)d_in[5];
    const float* bnb  = (const float*)d_in[6];
    const float* bnm  = (const float*)d_in[7];
    const float* bnv  = (const float*)d_in[8];
    const float* Wli1 = (const float*)d_in[9];
    const float* bli1 = (const float*)d_in[10];
    const float* p1   = (const float*)d_in[11];
    const float* Wl   = (const float*)d_in[12];
    const float* Wr   = (const float*)d_in[13];
    const float* att  = (const float*)d_in[14];
    const float* bgat = (const float*)d_in[15];
    const float* p2   = (const float*)d_in[16];
    const float* Wli2 = (const float*)d_in[17];
    const float* bli2 = (const float*)d_in[18];
    const float* Wli3 = (const float*)d_in[19];
    const float* bli3 = (const float*)d_in[20];
    float* out = (float*)d_out;

    char* ws = (char*)d_ws;
    size_t o = 0;
    auto take = [&](size_t bytes) -> char* {
        char* p = ws + o;
        o += (bytes + 255) & ~(size_t)255;
        return p;
    };
    float* deg   = (float*)take((size_t)N_TOT * 4);
    float* regA  = (float*)take((size_t)N_TOT * HID * 4);   // xw -> h1 -> xl|xr -> h3|h4
    float* regB  = (float*)take((size_t)N_TOT * HID * 4);   // h  -> hp -> h2
    float* enorm = (float*)take((size_t)EDGES * 4);         // edge norm -> ee/ex
    int*   newId = (int*)  take((size_t)N_TOT * 4);
    int*   s2    = (int*)  take((size_t)EDGES * 4);
    int*   d2    = (int*)  take((size_t)EDGES * 4);
    float* es    = (float*)take((size_t)NODES2 * 4);
    float* mbuf  = (float*)take((size_t)NODES2 * 4);
    float* ssum  = (float*)take((size_t)NODES2 * 4);

    float* xw = regA;
    float* h  = regB;
    float* h1 = regA;                          // xw dead after scatter
    float* hp = regB;                          // h dead after lin1 GEMM
    float* xl = regA;                          // h1 dead after pooling
    float* xr = regA + (size_t)NODES2 * HID;
    float* h2 = regB;                          // hp dead after xl/xr GEMM
    float* h3 = regA;                          // xl dead after GAT apply
    float* h4 = regA + (size_t)NODES3 * HID;

    // ---- GCNConv ----
    k_fill1        <<<N_TOT / 256, 256, 0, stream>>>(deg, N_TOT);
    k_deg_scatter  <<<EDGES / 256, 256, 0, stream>>>(dst, deg);
    k_rsqrt_inplace<<<N_TOT / 256, 256, 0, stream>>>(deg, N_TOT);
    k_gemm64<false, false><<<N_TOT / 64, 128, 0, stream>>>(
        x, W1, nullptr, xw, nullptr, nullptr, N_TOT,
        nullptr, nullptr, nullptr, nullptr, nullptr);
    k_edge_norm    <<<EDGES / 256, 256, 0, stream>>>(src, dst, deg, enorm);
    k_self_init    <<<(N_TOT * HID) / 256, 256, 0, stream>>>(xw, deg, h);
    k_gcn_scatter  <<<((size_t)EDGES * HID) / 256, 256, 0, stream>>>(src, dst, enorm, xw, h);

    // ---- BN (+b1) + LeakyReLU fused into lin1 WMMA GEMM ----
    k_gemm64<false, true><<<N_TOT / 64, 128, 0, stream>>>(
        h, Wli1, bli1, h1, nullptr, nullptr, N_TOT,
        b1, bng, bnb, bnm, bnv);

    // ---- TopKPooling 1 ----
    k_topk<<<NB, 256, 0, stream>>>(h1, p1, hp, newId, N_PER, KEEP1);

    // ---- GATv2 ----
    k_gemm64<true, false><<<NODES2 / 64, 128, 0, stream>>>(
        hp, Wl, nullptr, xl, Wr, xr, NODES2,
        nullptr, nullptr, nullptr, nullptr, nullptr);
    k_gat_self     <<<NODES2 / 256, 256, 0, stream>>>(xl, xr, att, es, mbuf);
    k_gat_edge_max <<<EDGES / 256, 256, 0, stream>>>(src, dst, newId, xl, xr, att, s2, d2, enorm, mbuf);
    k_gat_selfsum  <<<NODES2 / 256, 256, 0, stream>>>(es, mbuf, ssum);
    k_gat_edge_exp <<<EDGES / 256, 256, 0, stream>>>(d2, enorm, mbuf, ssum);
    k_gat_selfapply<<<(NODES2 * HID) / 256, 256, 0, stream>>>(es, mbuf, ssum, xl, h2);
    k_gat_edge_apply<<<((size_t)EDGES * HID) / 256, 256, 0, stream>>>(s2, d2, enorm, ssum, xl, h2);
    k_gat_final    <<<(NODES2 * HID) / 256, 256, 0, stream>>>(h2, bgat);

    // ---- TopKPooling 2 + lin2 ----
    k_topk<<<NB, 256, 0, stream>>>(h2, p2, h3, nullptr, KEEP1, KEEP2);
    k_gemm64<false, false><<<NODES3 / 64, 128, 0, stream>>>(
        h3, Wli2, bli2, h4, nullptr, nullptr, NODES3,
        nullptr, nullptr, nullptr, nullptr, nullptr);

    // ---- global_add_pool + head ----
    k_head<<<NB, 64, 0, stream>>>(h4, Wli3, bli3, out, KEEP2);
}